// Weights_model_12438225289500
// MI455X (gfx1250) — compile-verified
//
#include <hip/hip_runtime.h>
#include <math.h>

// Batched 100x100 slogdet via blocked LU with partial pivoting, one workgroup
// per frequency block. Matrix padded to 112x112 with identity (det-preserving)
// so the trailing update tiles as a 7x7 grid of 16x16 WMMA f32 tiles.
// Trailing update A22 -= L21*U12 uses V_WMMA_F32_16X16X4_F32 (the only f32
// WMMA on CDNA5; logdet needs f32 precision). Panel width = 4 == WMMA K.
// Wave id is read through readfirstlane so the WMMA tile loop is scalar-
// controlled and EXEC stays all-1s at every WMMA (ISA requirement).

#define BLOCKR 100
#define NP 112
#define LDA 113     // odd stride -> conflict-free column walks on 64 banks
#define NT 256      // 8 wave32 waves per workgroup

typedef __attribute__((ext_vector_type(2))) float v2f;
typedef __attribute__((ext_vector_type(8))) float v8f;

__global__ __launch_bounds__(NT)
void lu_logdet_kernel(const float* __restrict__ w,
                      const float* __restrict__ Lk,
                      float* __restrict__ out,
                      int n) {
  __shared__ float A[NP * LDA];      // ~50.6 KB, fits easily in 320KB WGP LDS
  __shared__ float ssig[BLOCKR];
  __shared__ float red_val[128];
  __shared__ int   red_idx[128];
  __shared__ int   s_p;
  __shared__ float s_inv;
  __shared__ float s_log;
  __shared__ int   s_sgn;

  const int f   = blockIdx.x;
  const int tid = threadIdx.x;

  if (tid < BLOCKR) ssig[tid] = 1.0f / (1.0f + expf(-w[f * BLOCKR + tid]));
  if (tid == 0) { s_log = 0.0f; s_sgn = 1; }
  __syncthreads();

  // Build padded M = s*(D - I) + I, identity in the 100..111 border.
  const long base = (long)(f * BLOCKR) * n + (long)(f * BLOCKR);
  for (int idx = tid; idx < NP * NP; idx += NT) {
    const int i = idx / NP, j = idx % NP;
    float v;
    if (i < BLOCKR && j < BLOCKR) {
      const float a = Lk[base + (long)i * n + j];
      v = ssig[i] * a;
      if (i == j) v += 1.0f - ssig[i];
    } else {
      v = (i == j) ? 1.0f : 0.0f;
    }
    A[i * LDA + j] = v;
  }
  __syncthreads();

  const int lane = tid & 31;
  // Scalarized wave id: guarantees the WMMA tile loop below is SGPR-controlled
  // so EXEC remains all-1s at the WMMA (required by the ISA).
  const int wv   = __builtin_amdgcn_readfirstlane(tid >> 5);
  const int half = lane >> 4;    // 0: lanes 0-15, 1: lanes 16-31
  const int lr   = lane & 15;

  for (int k = 0; k < NP / 4; ++k) {
    const int c0   = 4 * k;
    const int pend = c0 + 3;
    const int rs   = c0 + 4;     // start of trailing submatrix

    // ---------- panel factorization (columns c0..pend), partial pivoting ----
    for (int j = c0; j <= pend; ++j) {
      if (tid < 128) {
        const int r = j + tid;
        red_val[tid] = (r < NP) ? fabsf(A[r * LDA + j]) : -1.0f;
        red_idx[tid] = r;
      }
      for (int sft = 64; sft > 0; sft >>= 1) {
        __syncthreads();
        if (tid < sft && red_val[tid + sft] > red_val[tid]) {
          red_val[tid] = red_val[tid + sft];
          red_idx[tid] = red_idx[tid + sft];
        }
      }
      if (tid == 0) {
        const int p = red_idx[0];
        const float piv = A[p * LDA + j];
        s_p   = p;
        s_inv = 1.0f / piv;
        s_log += logf(fabsf(piv));
        int sg = (piv < 0.0f) ? -1 : 1;
        if (p != j) sg = -sg;
        s_sgn *= sg;
      }
      __syncthreads();
      // full row swap (including trailing columns)
      if (s_p != j) {
        const int p = s_p;
        for (int c = tid; c < NP; c += NT) {
          const float t0 = A[j * LDA + c];
          A[j * LDA + c] = A[p * LDA + c];
          A[p * LDA + c] = t0;
        }
      }
      __syncthreads();
      // scale multipliers
      {
        const int r = j + 1 + tid;
        if (r < NP) A[r * LDA + j] *= s_inv;
      }
      __syncthreads();
      // in-panel rank-1 update (columns j+1..pend only)
      const int nc = pend - j;
      if (nc > 0) {
        const int nr = NP - 1 - j;
        for (int idx = tid; idx < nr * nc; idx += NT) {
          const int r = j + 1 + idx / nc;
          const int c = j + 1 + idx % nc;
          A[r * LDA + c] -= A[r * LDA + j] * A[j * LDA + c];
        }
      }
      __syncthreads();
    }

    if (rs < NP) {
      // ---------- U12 = L11^{-1} * A12 (per-column, no barriers needed) ----
      for (int c = rs + tid; c < NP; c += NT) {
        for (int jj = c0 + 1; jj <= pend; ++jj) {
          float acc = A[jj * LDA + c];
          for (int m = c0; m < jj; ++m) acc -= A[jj * LDA + m] * A[m * LDA + c];
          A[jj * LDA + c] = acc;
        }
      }
      __syncthreads();

      // ---------- trailing update A22 -= L21 * U12 via WMMA f32 16x16x4 ----
      const int ts     = rs >> 4;
      const int dim    = (NP / 16) - ts;
      const int ntiles = dim * dim;          // uniform; loop control is scalar
      for (int t = wv; t < ntiles; t += NT / 32) {
        const int ti = __builtin_amdgcn_readfirstlane(ts + t / dim);
        const int tj = __builtin_amdgcn_readfirstlane(ts + t % dim);
        const int r0 = ti * 16;
        const int j0 = tj * 16;

        // A operand = -L21 (negated: f32 WMMA has no A/B NEG), masked rows<rs
        const float ma = (r0 + lr >= rs) ? -1.0f : 0.0f;
        v2f av, bv;
        av.x = ma * A[(r0 + lr) * LDA + c0 + 2 * half + 0];
        av.y = ma * A[(r0 + lr) * LDA + c0 + 2 * half + 1];
        // B operand = U12, masked cols<rs
        const float mb = (j0 + lr >= rs) ? 1.0f : 0.0f;
        bv.x = mb * A[(c0 + 2 * half + 0) * LDA + j0 + lr];
        bv.y = mb * A[(c0 + 2 * half + 1) * LDA + j0 + lr];
        // C tile (16x16 f32, VGPR v: row r0+v | r0+8+v)
        v8f cv;
#pragma unroll
        for (int v = 0; v < 8; ++v)
          cv[v] = A[(r0 + 8 * half + v) * LDA + j0 + lr];

        cv = __builtin_amdgcn_wmma_f32_16x16x4_f32(
            false, av, false, bv, (short)0, cv, false, false);

#pragma unroll
        for (int v = 0; v < 8; ++v)
          A[(r0 + 8 * half + v) * LDA + j0 + lr] = cv[v];
      }
    }
    __syncthreads();
  }

  if (tid == 0)
    out[1 + f] = (s_sgn > 0) ? s_log : __builtin_nanf("");
}

__global__ __launch_bounds__(128)
void sum_logdets_kernel(float* __restrict__ out, int nf) {
  __shared__ float sh[128];
  const int tid = threadIdx.x;
  sh[tid] = (tid < nf) ? out[1 + tid] : 0.0f;
  for (int sft = 64; sft > 0; sft >>= 1) {
    __syncthreads();
    if (tid < sft) sh[tid] += sh[tid + sft];
  }
  if (tid == 0) out[0] = sh[0];
}

extern "C" void kernel_launch(void* const* d_in, const int* in_sizes, int n_in,
                              void* d_out, int out_size, void* d_ws, size_t ws_size,
                              hipStream_t stream) {
  (void)n_in; (void)d_ws; (void)ws_size; (void)out_size;
  const float* w  = (const float*)d_in[0];
  const float* Lk = (const float*)d_in[1];
  float* out = (float*)d_out;
  const int n  = in_sizes[0];        // 12800
  const int nf = n / BLOCKR;         // 128

  lu_logdet_kernel<<<nf, NT, 0, stream>>>(w, Lk, out, n);
  sum_logdets_kernel<<<1, 128, 0, stream>>>(out, nf);
}